// Flattened_Align_90185723281676
// MI455X (gfx1250) — compile-verified
//
#include <hip/hip_runtime.h>
#include <cstdint>
#include <cstddef>

// ---------------------------------------------------------------------------
// MI455X (gfx1250) fused point-cloud losses.
//
// Roofline: ~232 MB streamed from HBM (@23.3 TB/s -> ~10us), gathers of the
// 24 MB xyz array are L2-resident (192 MB L2). Memory-bound -> one fused
// pass, coalesced b96/b128 loads, async global->LDS DMA for the KNN index
// tile (overlapped with per-point math), deterministic 2-stage reduction
// finished with V_WMMA_F32_16X16X4_F32 (ones-matrix column-sum reduction).
// ---------------------------------------------------------------------------

typedef float v2f __attribute__((ext_vector_type(2)));
typedef float v8f __attribute__((ext_vector_type(8)));

#define EPS_F 1e-8f
#define KNB 16
#define BLK 256
#define LDS_STRIDE 20  // ints per lane: 80B, 16B-aligned, reduces bank conflicts

struct F3 { float x, y, z; };

__global__ __launch_bounds__(BLK) void fused_point_kernel(
    const float* __restrict__ xyz,
    const float* __restrict__ scales,
    const float* __restrict__ rotation,
    const int*   __restrict__ knn,
    const float* __restrict__ normal,
    int N, float* __restrict__ partials)
{
  __shared__ __attribute__((aligned(16))) int s_knn[BLK * LDS_STRIDE];
  __shared__ float s_red[BLK / 32][3];

  const int t = threadIdx.x;
  const int i = blockIdx.x * BLK + t;
  const bool active = (i < N);

  float a_size = 0.f, a_var = 0.f, a_norm = 0.f;

  if (active) {
    // -------- stage this lane's 16 KNN indices (64 B) into LDS via async DMA
    uint32_t laddr = (uint32_t)(uintptr_t)(s_knn + t * LDS_STRIDE); // low 32b of flat LDS ptr == LDS offset
    uint64_t gaddr = (uint64_t)(uintptr_t)(knn + (size_t)i * KNB);
    asm volatile(
        "global_load_async_to_lds_b128 %0, %1, off\n\t"
        "global_load_async_to_lds_b128 %0, %1, off offset:16\n\t"
        "global_load_async_to_lds_b128 %0, %1, off offset:32\n\t"
        "global_load_async_to_lds_b128 %0, %1, off offset:48"
        :: "v"(laddr), "v"(gaddr) : "memory");

    // -------- overlapped per-point math while the DMA is in flight
    F3 sc = ((const F3*)scales)[i];
    float mins = sc.x; int jmin = 0;
    if (sc.y < mins) { mins = sc.y; jmin = 1; }
    if (sc.z < mins) { mins = sc.z; jmin = 2; }
    a_size = mins;

    F3 nr = ((const F3*)normal)[i];
    float inl = 1.f / (sqrtf(nr.x*nr.x + nr.y*nr.y + nr.z*nr.z) + EPS_F);
    float nx = nr.x * inl, ny = nr.y * inl, nz = nr.z * inl;

    float4 q = ((const float4*)rotation)[i];   // (w,x,y,z) = (q0,q1,q2,q3)
    float iq = 1.f / (sqrtf(q.x*q.x + q.y*q.y + q.z*q.z + q.w*q.w) + EPS_F);
    float qw = q.x*iq, qx = q.y*iq, qy = q.z*iq, qz = q.w*iq;

    float ax, ay, az;                          // column jmin of R(q)
    if (jmin == 0)      { ax = 1.f-2.f*(qy*qy+qz*qz); ay = 2.f*(qx*qy+qw*qz); az = 2.f*(qx*qz-qw*qy); }
    else if (jmin == 1) { ax = 2.f*(qx*qy-qw*qz); ay = 1.f-2.f*(qx*qx+qz*qz); az = 2.f*(qy*qz+qw*qx); }
    else                { ax = 2.f*(qx*qz+qw*qy); ay = 2.f*(qy*qz-qw*qx); az = 1.f-2.f*(qx*qx+qy*qy); }
    a_norm = 1.f - fabsf(ax*nx + ay*ny + az*nz);

    F3 p = ((const F3*)xyz)[i];
    float pd = p.x*nx + p.y*ny + p.z*nz;

    // -------- wait for index DMA, then do the L2-resident gathers
    asm volatile("s_wait_asynccnt 0" ::: "memory");

    float d[KNB];
    float sum = 0.f;
    #pragma unroll
    for (int k = 0; k < KNB; ++k) {
      int nb = s_knn[t * LDS_STRIDE + k];
      F3 pn = ((const F3*)xyz)[nb];
      d[k] = pn.x*nx + pn.y*ny + pn.z*nz - pd;
      sum += d[k];
    }
    float mean = sum * (1.f / 16.f);
    float var = 0.f;
    #pragma unroll
    for (int k = 0; k < KNB; ++k) { float e = d[k] - mean; var += e * e; }
    a_var = var;
  }

  // -------- deterministic in-block reduction (wave32 shuffles, then LDS)
  #pragma unroll
  for (int off = 16; off > 0; off >>= 1) {
    a_size += __shfl_down(a_size, off, 32);
    a_var  += __shfl_down(a_var,  off, 32);
    a_norm += __shfl_down(a_norm, off, 32);
  }
  const int wave = t >> 5, lane = t & 31;
  if (lane == 0) { s_red[wave][0] = a_size; s_red[wave][1] = a_var; s_red[wave][2] = a_norm; }
  __syncthreads();
  if (t == 0) {
    float b0 = 0.f, b1 = 0.f, b2 = 0.f;
    #pragma unroll
    for (int w = 0; w < BLK / 32; ++w) { b0 += s_red[w][0]; b1 += s_red[w][1]; b2 += s_red[w][2]; }
    partials[(size_t)blockIdx.x * 3 + 0] = b0;
    partials[(size_t)blockIdx.x * 3 + 1] = b1;
    partials[(size_t)blockIdx.x * 3 + 2] = b2;
  }
}

// ---------------------------------------------------------------------------
// Final reduction: one block. Thread-strided deterministic accumulation of
// block partials, then a V_WMMA_F32_16X16X4_F32 ones-matrix reduction:
// with A == all-ones, D[i][j] = sum_c B[c][j], so chaining 4 accumulating
// WMMAs over 4x64-value chunks yields 16 column partial sums covering all
// 256 values exactly once; a 16-lane xor-shuffle finishes the sum.
// ---------------------------------------------------------------------------
__global__ __launch_bounds__(BLK) void final_reduce_kernel(
    const float* __restrict__ partials, int nb, int N, float* __restrict__ out)
{
  __shared__ float s_c[3][BLK];
  const int t = threadIdx.x;
  float c0 = 0.f, c1 = 0.f, c2 = 0.f;
  for (int b = t; b < nb; b += BLK) {
    c0 += partials[(size_t)b * 3 + 0];
    c1 += partials[(size_t)b * 3 + 1];
    c2 += partials[(size_t)b * 3 + 2];
  }
  s_c[0][t] = c0; s_c[1][t] = c1; s_c[2][t] = c2;
  __syncthreads();

  if (t < 32) {                       // wave 0 only: EXEC all-ones for WMMA
    v2f ones; ones[0] = 1.f; ones[1] = 1.f;
    float res[3];
    #pragma unroll
    for (int c = 0; c < 3; ++c) {
      v8f acc = {};
      #pragma unroll
      for (int ch = 0; ch < 4; ++ch) {
        v2f b;
        b[0] = s_c[c][ch * 64 + t];
        b[1] = s_c[c][ch * 64 + 32 + t];
        acc = __builtin_amdgcn_wmma_f32_16x16x4_f32(
            /*neg_a=*/false, ones, /*neg_b=*/false, b,
            /*c_mod=*/(short)0, acc, /*reuse_a=*/false, /*reuse_b=*/false);
      }
      // acc[0] on lanes 0..15 = D[0][0..15] = the 16 column sums
      float v = acc[0];
      #pragma unroll
      for (int off = 8; off > 0; off >>= 1) v += __shfl_xor(v, off, 16);
      res[c] = v;
    }
    if (t == 0) {
      out[0] = res[0] / (float)N;                 // loss_size
      out[1] = res[1] / ((float)N * 16.f);        // loss_d
      out[2] = res[2] / (float)N;                 // loss_normal
    }
  }
}

extern "C" void kernel_launch(void* const* d_in, const int* in_sizes, int n_in,
                              void* d_out, int out_size, void* d_ws, size_t ws_size,
                              hipStream_t stream) {
  const float* xyz    = (const float*)d_in[0];
  // d_in[1] = xyz_id (unused by the reference math)
  const float* scales = (const float*)d_in[2];
  const float* rot    = (const float*)d_in[3];
  const int*   knn    = (const int*)d_in[4];
  const float* nrm    = (const float*)d_in[5];

  const int N  = in_sizes[0] / 3;
  const int nb = (N + BLK - 1) / BLK;
  float* partials = (float*)d_ws;    // nb*3 floats, fully rewritten each call

  fused_point_kernel<<<nb, BLK, 0, stream>>>(xyz, scales, rot, knn, nrm, N, partials);
  final_reduce_kernel<<<1, BLK, 0, stream>>>(partials, nb, N, (float*)d_out);
}